// MLPs_90752658965186
// MI455X (gfx1250) — compile-verified
//
#include <hip/hip_runtime.h>
#include <math.h>

// ---------------------------------------------------------------------------
// Fused HGNN feature-extraction MLP stack for MI455X (gfx1250, wave32).
// All dense layers run on V_WMMA_F32_16X16X4_F32 (exact fp32, D = A*B + C).
// One wave = one 16-row tile of op-nodes, carried through all 9 linears.
// Activations ping-pong through per-wave LDS so A-operands can be gathered
// in the ISA 16x4 A-layout; weights (B-operands) stream from global (L2-hot).
// ELU is branchless via v_exp_f32 (TRANS pipe, co-issues around WMMA).
// ---------------------------------------------------------------------------

typedef __attribute__((ext_vector_type(2))) float v2f;
typedef __attribute__((ext_vector_type(4))) float v4f;
typedef __attribute__((ext_vector_type(8))) float v8f;

#define WAVES 4
#define TPB   (WAVES * 32)
#define BATCH 32
#define NOP   512
#define NMA   64
#define FDIM  8
#define HDIM  128
#define ODIM  8

// Branchless ELU: x>0 ? x : exp(x)-1, exp via single v_exp_f32 (exp2).
__device__ __forceinline__ float elu1(float x) {
  float e = __builtin_amdgcn_exp2f(x * 1.44269504088896340736f) - 1.0f;
  return x > 0.0f ? x : e;
}

__device__ __forceinline__ v8f wmma_f32(v2f a, v2f b, v8f c) {
  // (neg_a, A, neg_b, B, c_mod, C, reuse_a, reuse_b)
  return __builtin_amdgcn_wmma_f32_16x16x4_f32(false, a, false, b, (short)0, c,
                                               false, false);
}

// Y(16 x 128) = act(X(16 x K) @ W(K x 128) + bias).  X, Y in LDS (per-wave).
template <int K, bool ELU>
__device__ __forceinline__ void layer_h(const float* __restrict__ W,
                                        const float* __restrict__ bias,
                                        const float* sA, float* sY, int lane) {
  const int half = lane >> 4, l16 = lane & 15;
  const int kh = half * 2;
  for (int nt = 0; nt < HDIM / 16; ++nt) {
    const int n0 = nt * 16;
    v8f acc0 = {0.f, 0.f, 0.f, 0.f, 0.f, 0.f, 0.f, 0.f};
    v8f acc1 = {0.f, 0.f, 0.f, 0.f, 0.f, 0.f, 0.f, 0.f};
#pragma unroll
    for (int kc = 0; kc < K / 4; kc += 2) {
      {
        const int k = kc * 4 + kh;
        v2f a; a.x = sA[l16 * K + k];            a.y = sA[l16 * K + k + 1];
        v2f b; b.x = W[k * HDIM + n0 + l16];     b.y = W[(k + 1) * HDIM + n0 + l16];
        acc0 = wmma_f32(a, b, acc0);
      }
      {
        const int k = (kc + 1) * 4 + kh;
        v2f a; a.x = sA[l16 * K + k];            a.y = sA[l16 * K + k + 1];
        v2f b; b.x = W[k * HDIM + n0 + l16];     b.y = W[(k + 1) * HDIM + n0 + l16];
        acc1 = wmma_f32(a, b, acc1);
      }
    }
    const float bv = bias[n0 + l16];
#pragma unroll
    for (int r = 0; r < 8; ++r) {
      float y = acc0[r] + acc1[r] + bv;
      if (ELU) y = elu1(y);
      sY[(r + half * 8) * HDIM + n0 + l16] = y;  // D layout: M = r + 8*half, N = lane16
    }
  }
}

// Head: D(16 x 16) = X(16 x K) @ W(K x 8) + bias.  Cols 8..15 are dead lanes.
template <int K>
__device__ __forceinline__ v8f layer_o(const float* __restrict__ W,
                                       const float* __restrict__ bias,
                                       const float* sA, int lane) {
  const int half = lane >> 4, l16 = lane & 15;
  const int kh = half * 2;
  const int n = (l16 < 8) ? l16 : 0;  // clamp to stay in-bounds; cols>=8 unused
  v8f acc0 = {0.f, 0.f, 0.f, 0.f, 0.f, 0.f, 0.f, 0.f};
  v8f acc1 = {0.f, 0.f, 0.f, 0.f, 0.f, 0.f, 0.f, 0.f};
#pragma unroll
  for (int kc = 0; kc < K / 4; kc += 2) {
    {
      const int k = kc * 4 + kh;
      v2f a; a.x = sA[l16 * K + k];    a.y = sA[l16 * K + k + 1];
      v2f b; b.x = W[k * ODIM + n];    b.y = W[(k + 1) * ODIM + n];
      acc0 = wmma_f32(a, b, acc0);
    }
    {
      const int k = (kc + 1) * 4 + kh;
      v2f a; a.x = sA[l16 * K + k];    a.y = sA[l16 * K + k + 1];
      v2f b; b.x = W[k * ODIM + n];    b.y = W[(k + 1) * ODIM + n];
      acc1 = wmma_f32(a, b, acc1);
    }
  }
  const float bv = bias[n];
#pragma unroll
  for (int r = 0; r < 8; ++r) acc0[r] += acc1[r] + bv;
  return acc0;
}

__global__ __launch_bounds__(TPB) void fused_hgnn_mlp(
    const int* __restrict__ adj, const float* __restrict__ feats_op,
    const float* __restrict__ feats_ma,
    const float* __restrict__ l0w1, const float* __restrict__ l0b1,
    const float* __restrict__ l0w2, const float* __restrict__ l0b2,
    const float* __restrict__ l0w3, const float* __restrict__ l0b3,
    const float* __restrict__ l1w1, const float* __restrict__ l1b1,
    const float* __restrict__ l1w2, const float* __restrict__ l1b2,
    const float* __restrict__ l1w3, const float* __restrict__ l1b3,
    const float* __restrict__ pw1, const float* __restrict__ pb1,
    const float* __restrict__ pw2, const float* __restrict__ pb2,
    const float* __restrict__ pw3, const float* __restrict__ pb3,
    float* __restrict__ out) {
  __shared__ float sFM[NMA * FDIM];            // feats_ma for this batch (2 KB)
  __shared__ float sBuf[WAVES][2][16 * HDIM];  // per-wave activation ping-pong
  __shared__ float sCat[WAVES][16 * 16];       // elu(concat(emb0, emb1))

  const int tid = threadIdx.x;
  const int wid = tid >> 5, lane = tid & 31;
  const int half = lane >> 4, l16 = lane & 15;
  const int row0 = blockIdx.x * (WAVES * 16) + wid * 16;  // global row of tile
  const int bb = row0 / NOP;
  const int op0 = row0 % NOP;

  // Warm WGP$/L2 with the three 64KB hidden weight matrices (global_prefetch).
  __builtin_prefetch(l0w2 + tid * 128, 0, 1);
  __builtin_prefetch(l1w2 + tid * 128, 0, 1);
  __builtin_prefetch(pw2 + tid * 128, 0, 1);

  // Stage feats_ma (batch-shared) into LDS.
  for (int i = tid; i < NMA * FDIM; i += TPB)
    sFM[i] = feats_ma[bb * NMA * FDIM + i];
  __syncthreads();

  float* sA = &sBuf[wid][0][0];
  float* sB = &sBuf[wid][1][0];
  float* sC = &sCat[wid][0];

  // ---- branch 0: agg = adj_tile(16x64) @ feats_ma(64x8)  (VALU, 0/1 weights)
  {
    const int m = lane >> 1, k0 = (lane & 1) * 4;  // 4 outputs per lane
    const int* arow = adj + ((size_t)bb * NOP + op0 + m) * NMA;
    v4f s = {0.f, 0.f, 0.f, 0.f};
    for (int j = 0; j < NMA; ++j) {
      const float av = (float)arow[j];
      const float* f = &sFM[j * FDIM + k0];
      s.x += av * f[0]; s.y += av * f[1]; s.z += av * f[2]; s.w += av * f[3];
    }
    *(v4f*)&sA[m * FDIM + k0] = s;
  }
  __syncthreads();
  layer_h<FDIM, true>(l0w1, l0b1, sA, sB, lane);
  __syncthreads();
  layer_h<HDIM, true>(l0w2, l0b2, sB, sA, lane);
  __syncthreads();
  {
    v8f e0 = layer_o<HDIM>(l0w3, l0b3, sA, lane);
    if (l16 < 8) {
#pragma unroll
      for (int r = 0; r < 8; ++r)
        sC[(r + half * 8) * 16 + l16] = elu1(e0[r]);  // elu(cat), low half
    }
  }
  __syncthreads();

  // ---- branch 1: self-loop path, X = feats_op rows
  {
    const int m = lane >> 1, k0 = (lane & 1) * 4;
    const float* fr = feats_op + ((size_t)bb * NOP + op0 + m) * FDIM + k0;
    *(v4f*)&sA[m * FDIM + k0] = *(const v4f*)fr;
  }
  __syncthreads();
  layer_h<FDIM, true>(l1w1, l1b1, sA, sB, lane);
  __syncthreads();
  layer_h<HDIM, true>(l1w2, l1b2, sB, sA, lane);
  __syncthreads();
  {
    v8f e1 = layer_o<HDIM>(l1w3, l1b3, sA, lane);
    if (l16 < 8) {
#pragma unroll
      for (int r = 0; r < 8; ++r)
        sC[(r + half * 8) * 16 + 8 + l16] = elu1(e1[r]);  // elu(cat), high half
    }
  }
  __syncthreads();

  // ---- final projection on elu(cat) (K = 16)
  layer_h<16, true>(pw1, pb1, sC, sA, lane);
  __syncthreads();
  layer_h<HDIM, true>(pw2, pb2, sA, sB, lane);
  __syncthreads();
  {
    v8f mu = layer_o<HDIM>(pw3, pb3, sB, lane);
    if (l16 < 8) {
#pragma unroll
      for (int r = 0; r < 8; ++r)
        out[(size_t)(row0 + r + half * 8) * ODIM + l16] = mu[r];
    }
  }
}

extern "C" void kernel_launch(void* const* d_in, const int* in_sizes, int n_in,
                              void* d_out, int out_size, void* d_ws,
                              size_t ws_size, hipStream_t stream) {
  const int*   adj = (const int*)d_in[0];
  const float* fop = (const float*)d_in[1];
  const float* fma = (const float*)d_in[2];
  const float* l0w1 = (const float*)d_in[3];
  const float* l0b1 = (const float*)d_in[4];
  const float* l0w2 = (const float*)d_in[5];
  const float* l0b2 = (const float*)d_in[6];
  const float* l0w3 = (const float*)d_in[7];
  const float* l0b3 = (const float*)d_in[8];
  const float* l1w1 = (const float*)d_in[9];
  const float* l1b1 = (const float*)d_in[10];
  const float* l1w2 = (const float*)d_in[11];
  const float* l1b2 = (const float*)d_in[12];
  const float* l1w3 = (const float*)d_in[13];
  const float* l1b3 = (const float*)d_in[14];
  const float* pw1 = (const float*)d_in[15];
  const float* pb1 = (const float*)d_in[16];
  const float* pw2 = (const float*)d_in[17];
  const float* pb2 = (const float*)d_in[18];
  const float* pw3 = (const float*)d_in[19];
  const float* pb3 = (const float*)d_in[20];

  const int total_rows = BATCH * NOP;            // 16384
  const int rows_per_block = WAVES * 16;         // 64
  dim3 grid(total_rows / rows_per_block);        // 256 blocks
  fused_hgnn_mlp<<<grid, TPB, 0, stream>>>(
      adj, fop, fma, l0w1, l0b1, l0w2, l0b2, l0w3, l0b3, l1w1, l1b1, l1w2,
      l1b2, l1w3, l1b3, pw1, pb1, pw2, pb2, pw3, pb3, (float*)d_out);
}